// MPNN_79628693668166
// MI455X (gfx1250) — compile-verified
//
#include <hip/hip_runtime.h>
#include <hip/hip_bf16.h>

#define N_NODES   100000
#define N_EDGES   1600000
#define HID       128
#define NGR       64
#define NCLS      10

typedef __attribute__((ext_vector_type(2))) float v2f;
typedef __attribute__((ext_vector_type(8))) float v8f;

// ---------------------------------------------------------------------------
// Zero a float buffer (grid-stride)
// ---------------------------------------------------------------------------
__global__ void zero_f32(float* __restrict__ p, size_t n) {
    size_t i = (size_t)blockIdx.x * blockDim.x + threadIdx.x;
    size_t stride = (size_t)gridDim.x * blockDim.x;
    for (; i < n; i += stride) p[i] = 0.0f;
}

// ---------------------------------------------------------------------------
// Edge scatter-add: agg[dst] += h[src].  One wave (32 lanes) per edge; each
// lane handles 4 contiguous features via a float4 gather (coalesced 512B per
// wave) followed by 4 f32 atomics.  agg (51.2MB) is L2-resident -> atomics
// resolve at L2.
// ---------------------------------------------------------------------------
__global__ void scatter_add(const float* __restrict__ hin,
                            const int* __restrict__ ei,
                            float* __restrict__ agg) {
    int gid  = blockIdx.x * blockDim.x + threadIdx.x;
    int e    = gid >> 5;
    int lane = gid & 31;
    if (e >= N_EDGES) return;
    int src = ei[e];
    int dst = ei[N_EDGES + e];
    const float4 v = *(const float4*)(hin + (size_t)src * HID + lane * 4);
    float* a = agg + (size_t)dst * HID + lane * 4;
    atomicAdd(a + 0, v.x);
    atomicAdd(a + 1, v.y);
    atomicAdd(a + 2, v.z);
    atomicAdd(a + 3, v.w);
}

// ---------------------------------------------------------------------------
// Fused SAGE GEMM:  hout = [relu]( agg @ Wl^T + hin @ Wr^T + b )
// Block = 256 threads = 8 waves.  Block b owns node rows [16b, 16b+16);
// wave w owns output columns [16w, 16w+16).  Each wave runs 64
// V_WMMA_F32_16X16X4_F32 (K=128, two fused matmuls) into one f32 accum.
//
// A layout (16x4 f32, ISA 7.12.2): lane m=lane%16 holds K = k0,k0+1 in
// VGPR0,1 with k0=(lane/16)*2  -> contiguous float2 from the node row.
// B layout (4x16): lane n=lane%16 holds K = k0,k0+1 of column jbase+n,
// i.e. Wl[jbase+n][k0..k0+1] -> contiguous float2 from the weight row
// (B = W^T exactly as `@ W.T` requires).
// C/D layout: VGPR r -> row r (lanes 0-15) / row r+8 (lanes 16-31), n=lane%16.
// ---------------------------------------------------------------------------
__global__ void sage_gemm(const float* __restrict__ agg,
                          const float* __restrict__ hin,
                          const float* __restrict__ Wl,
                          const float* __restrict__ Wr,
                          const float* __restrict__ bias,
                          float* __restrict__ hout,
                          int relu) {
    const int lane = threadIdx.x & 31;
    const int wave = threadIdx.x >> 5;
    const int ibase = blockIdx.x * 16;   // node rows
    const int jbase = wave * 16;         // output cols

    const int p16  = lane & 15;          // m for A-rows, n for B/D-cols
    const int kgrp = (lane >> 4) * 2;    // 0 or 2

    const float* arow = agg + (size_t)(ibase + p16) * HID;
    const float* hrow = hin + (size_t)(ibase + p16) * HID;
    const float* lrow = Wl  + (size_t)(jbase + p16) * HID;
    const float* rrow = Wr  + (size_t)(jbase + p16) * HID;

    v8f c = {0.f, 0.f, 0.f, 0.f, 0.f, 0.f, 0.f, 0.f};

#pragma unroll 4
    for (int kk = 0; kk < HID; kk += 4) {
        const int k = kk + kgrp;
        v2f aA = *(const v2f*)(arow + k);
        v2f aH = *(const v2f*)(hrow + k);
        v2f bL = *(const v2f*)(lrow + k);
        v2f bR = *(const v2f*)(rrow + k);
        // 8-arg pattern: (neg_a, A, neg_b, B, c_mod, C, reuse_a, reuse_b)
        c = __builtin_amdgcn_wmma_f32_16x16x4_f32(false, aA, false, bL,
                                                  (short)0, c, false, false);
        c = __builtin_amdgcn_wmma_f32_16x16x4_f32(false, aH, false, bR,
                                                  (short)0, c, false, false);
    }

    const float bj = bias[jbase + p16];
    const int mb = ibase + ((lane >> 4) << 3);   // +0 or +8
    const int nj = jbase + p16;
#pragma unroll
    for (int r = 0; r < 8; ++r) {
        float v = c[r] + bj;
        if (relu) v = fmaxf(v, 0.0f);
        hout[(size_t)(mb + r) * HID + nj] = v;
    }
}

// ---------------------------------------------------------------------------
// Graph pooling: pooled[batch[n]] += h[n]  (atomic segment-sum)
// ---------------------------------------------------------------------------
__global__ void pool_sum(const float* __restrict__ h,
                         const int* __restrict__ batch,
                         float* __restrict__ pooled) {
    int gid = blockIdx.x * blockDim.x + threadIdx.x;
    if (gid >= N_NODES * HID) return;
    int n = gid >> 7;
    int f = gid & 127;
    atomicAdd(&pooled[(size_t)batch[n] * HID + f], h[gid]);
}

// ---------------------------------------------------------------------------
// Classifier head: out[g][j] = pooled[g] . Wout[j] + bout[j]   (64x10, tiny)
// ---------------------------------------------------------------------------
__global__ void head_gemm(const float* __restrict__ pooled,
                          const float* __restrict__ Wout,
                          const float* __restrict__ bout,
                          float* __restrict__ out) {
    int t = threadIdx.x;
    if (t >= NGR * NCLS) return;
    int g = t / NCLS, j = t % NCLS;
    const float* p = pooled + (size_t)g * HID;
    const float* w = Wout + (size_t)j * HID;
    float s = bout[j];
#pragma unroll 8
    for (int k = 0; k < HID; ++k) s += p[k] * w[k];
    out[t] = s;
}

// ---------------------------------------------------------------------------
extern "C" void kernel_launch(void* const* d_in, const int* in_sizes, int n_in,
                              void* d_out, int out_size, void* d_ws, size_t ws_size,
                              hipStream_t stream) {
    const float* x     = (const float*)d_in[0];
    const int*   ei    = (const int*)d_in[1];
    const int*   batch = (const int*)d_in[2];
    const float* W1l   = (const float*)d_in[3];
    const float* b1    = (const float*)d_in[4];
    const float* W1r   = (const float*)d_in[5];
    const float* W2l   = (const float*)d_in[6];
    const float* b2    = (const float*)d_in[7];
    const float* W2r   = (const float*)d_in[8];
    const float* W3l   = (const float*)d_in[9];
    const float* b3    = (const float*)d_in[10];
    const float* W3r   = (const float*)d_in[11];
    const float* Wout  = (const float*)d_in[12];
    const float* bout  = (const float*)d_in[13];
    float* out = (float*)d_out;

    const size_t feat_elems = (size_t)N_NODES * HID;   // 12.8M floats
    float* agg    = (float*)d_ws;
    float* bufA   = agg  + feat_elems;
    float* bufB   = bufA + feat_elems;
    float* pooled = bufB + feat_elems;                 // 64*128 floats

    const int scatter_blocks = (N_EDGES * 32) / 256;   // exact: 200000
    const int gemm_blocks    = N_NODES / 16;           // exact: 6250
    const int pool_blocks    = (N_NODES * HID) / 256;  // exact: 50000

    auto layer = [&](const float* hin, const float* Wl, const float* b,
                     const float* Wr, float* hout, int relu) {
        zero_f32<<<2048, 256, 0, stream>>>(agg, feat_elems);
        scatter_add<<<scatter_blocks, 256, 0, stream>>>(hin, ei, agg);
        sage_gemm<<<gemm_blocks, 256, 0, stream>>>(agg, hin, Wl, Wr, b, hout, relu);
    };

    layer(x,    W1l, b1, W1r, bufA, 1);
    layer(bufA, W2l, b2, W2r, bufB, 1);
    layer(bufB, W3l, b3, W3r, bufA, 0);

    zero_f32<<<64, 256, 0, stream>>>(pooled, (size_t)NGR * HID);
    pool_sum<<<pool_blocks, 256, 0, stream>>>(bufA, batch, pooled);
    head_gemm<<<1, NGR * NCLS, 0, stream>>>(pooled, Wout, bout, out);
}